// RNN_755914244859
// MI455X (gfx1250) — compile-verified
//
#include <hip/hip_runtime.h>
#include <math.h>

// GRU scan for MI455X (gfx1250, wave32) — single persistent kernel:
//  - fp32 WMMA 16x16x4 for both x-side and h-side gate GEMMs
//  - 32 blocks loop over all 512 time steps; device-wide barrier between
//    steps via monotonic global atomic counter (agent scope, s_sleep backoff)
//    -> no per-step kernel-launch overhead on the sequential critical path
//  - per-block LDS staging of the 16x512 x-tile and h-tile (shared by all
//    8 waves); 516-float row stride -> bank-conflict-free fragment reads
//  - global_prefetch of next step's x-tile to hide HBM latency
//  - weights (6 MB) stay resident in 192 MB L2 across all 512 steps;
//    weight pointers & biases hoisted out of the time loop

typedef float v2f __attribute__((ext_vector_type(2)));
typedef float v8f __attribute__((ext_vector_type(8)));

#define BATCH 128
#define SEQ   512
#define ISZ   512
#define HSZ   512
#define LDST  516   // padded LDS row stride (floats): banks 4m+c all distinct
#define NBLK  32

__global__ void gru_init_kernel(float* __restrict__ h0,
                                unsigned* __restrict__ counter, int n) {
  int i = blockIdx.x * blockDim.x + threadIdx.x;
  if (i < n) h0[i] = 0.0f;
  if (i == 0) *counter = 0u;
}

__device__ __forceinline__ v2f ld2g(const float* __restrict__ p) {
  return *(const v2f*)p;
}

__device__ __forceinline__ float sigmoidf_(float x) {
  return 1.0f / (1.0f + expf(-x));
}

// Grid: 32 blocks x 256 threads (8 waves).
// Block: row_tile = blockIdx>>2 (16 batch rows), colgroup = blockIdx&3.
// Wave w handles col_tile = colgroup*8 + w -> 16 H columns.
// Accumulators per wave: r, z (x+h summed), n_x, n_h -> 4x v8f (32 VGPRs).
__global__ __launch_bounds__(256) void gru_persistent_kernel(
    const float* __restrict__ xs,      // [B, T, I]
    const float* __restrict__ wih,     // [3H, I]
    const float* __restrict__ whh,     // [3H, H]
    const float* __restrict__ bias,    // [3H]
    const float* __restrict__ bias_n,  // [H]
    float* __restrict__ hA,            // [B, H] ping
    float* __restrict__ hB,            // [B, H] pong
    unsigned* __restrict__ counter,    // grid barrier (monotonic)
    float* __restrict__ out)           // [B, H] final h
{
  __shared__ float lds_x[16 * LDST];
  __shared__ float lds_h[16 * LDST];

  const int tid      = threadIdx.x;
  const int w        = tid >> 5;
  const int lane     = tid & 31;
  const int row_tile = blockIdx.x >> 2;                 // 0..7
  const int col_tile = (blockIdx.x & 3) * 8 + w;        // 0..31
  const int rowbase  = row_tile * 16;
  const int colbase  = col_tile * 16;

  const int m    = lane & 15;            // M index (A) / N index (B,C,D)
  const int half = lane >> 4;            // K-pair selector for A/B frags

  // Invariant over t: weight fragment base pointers and biases.
  const float* WIr = wih + (size_t)(0 * HSZ + colbase + m) * ISZ + 2 * half;
  const float* WIz = wih + (size_t)(1 * HSZ + colbase + m) * ISZ + 2 * half;
  const float* WIn = wih + (size_t)(2 * HSZ + colbase + m) * ISZ + 2 * half;
  const float* WHr = whh + (size_t)(0 * HSZ + colbase + m) * HSZ + 2 * half;
  const float* WHz = whh + (size_t)(1 * HSZ + colbase + m) * HSZ + 2 * half;
  const float* WHn = whh + (size_t)(2 * HSZ + colbase + m) * HSZ + 2 * half;

  const int   c0  = colbase + m;         // this lane's output column
  const float br  = bias[c0];
  const float bz  = bias[HSZ + c0];
  const float bnx = bias[2 * HSZ + c0];
  const float bn  = bias_n[c0];

  for (int t = 0; t < SEQ; ++t) {
    const float* h_in  = (t & 1) ? hB : hA;
    float*       h_out = (t == SEQ - 1) ? out : ((t & 1) ? hA : hB);

    // ---- cooperative stage of A tiles: x_t[16 x 512] and h[16 x 512] ----
    {
      const float* xbase = xs + (size_t)rowbase * SEQ * ISZ + (size_t)t * ISZ;
#pragma unroll
      for (int c = 0; c < 8; ++c) {
        int idx  = c * 256 + tid;        // float4 index; 2048 total
        int row  = idx >> 7;             // 128 float4 per 512-float row
        int col4 = (idx & 127) * 4;
        float4 vx = *(const float4*)(xbase + (size_t)row * SEQ * ISZ + col4);
        float4 vh = *(const float4*)(h_in + (size_t)(rowbase + row) * HSZ + col4);
        *(float4*)(&lds_x[row * LDST + col4]) = vx;
        *(float4*)(&lds_h[row * LDST + col4]) = vh;
      }
    }

    // Prefetch next step's x-tile into cache (off the critical path).
    if (t + 1 < SEQ) {
      const float* xn = xs + (size_t)(rowbase + (tid >> 4)) * SEQ * ISZ
                          + (size_t)(t + 1) * ISZ + (tid & 15) * 32;
      __builtin_prefetch(xn, 0, 1);
    }

    __syncthreads();

    v8f ar = {}, az = {}, anx = {}, anh = {};

    // -------- x-side GEMM: A from LDS, K = ISZ --------
    {
      const float* A = &lds_x[m * LDST + 2 * half];
      for (int k = 0; k < ISZ; k += 4) {
        v2f a = *(const v2f*)(A + k);
        ar  = __builtin_amdgcn_wmma_f32_16x16x4_f32(false, a, false, ld2g(WIr + k), (short)0, ar,  false, false);
        az  = __builtin_amdgcn_wmma_f32_16x16x4_f32(false, a, false, ld2g(WIz + k), (short)0, az,  false, false);
        anx = __builtin_amdgcn_wmma_f32_16x16x4_f32(false, a, false, ld2g(WIn + k), (short)0, anx, false, false);
      }
    }

    // -------- h-side GEMM: A from LDS, K = HSZ --------
    {
      const float* A = &lds_h[m * LDST + 2 * half];
      for (int k = 0; k < HSZ; k += 4) {
        v2f a = *(const v2f*)(A + k);
        ar  = __builtin_amdgcn_wmma_f32_16x16x4_f32(false, a, false, ld2g(WHr + k), (short)0, ar,  false, false);
        az  = __builtin_amdgcn_wmma_f32_16x16x4_f32(false, a, false, ld2g(WHz + k), (short)0, az,  false, false);
        anh = __builtin_amdgcn_wmma_f32_16x16x4_f32(false, a, false, ld2g(WHn + k), (short)0, anh, false, false);
      }
    }

    // -------- GRU nonlinearity + state update --------
    // C/D layout: VGPR v, lane -> row = rowbase + v + 8*half, col = c0
#pragma unroll
    for (int v = 0; v < 8; ++v) {
      const int rl = v + 8 * half;                // local row 0..15
      float r = sigmoidf_(ar[v] + br);
      float z = sigmoidf_(az[v] + bz);
      float n = tanhf(anx[v] + bnx + r * (anh[v] + bn));
      float h = lds_h[rl * LDST + c0];
      h_out[(size_t)(rowbase + rl) * HSZ + c0] = n + z * (h - n);
    }

    // -------- device-wide barrier between steps --------
    if (t + 1 < SEQ) {
      __threadfence();                            // publish h_out stores
      __syncthreads();
      if (tid == 0) {
        __hip_atomic_fetch_add(counter, 1u, __ATOMIC_RELEASE,
                               __HIP_MEMORY_SCOPE_AGENT);
        const unsigned want = (unsigned)NBLK * (unsigned)(t + 1);
        while (__hip_atomic_load(counter, __ATOMIC_ACQUIRE,
                                 __HIP_MEMORY_SCOPE_AGENT) < want) {
          __builtin_amdgcn_s_sleep(2);
        }
      }
      __syncthreads();
    }
  }
}

extern "C" void kernel_launch(void* const* d_in, const int* in_sizes, int n_in,
                              void* d_out, int out_size, void* d_ws, size_t ws_size,
                              hipStream_t stream) {
  (void)in_sizes; (void)n_in; (void)out_size; (void)ws_size;

  const float* xs     = (const float*)d_in[0];  // [128, 512, 512]
  const float* wih    = (const float*)d_in[1];  // [1536, 512]
  const float* whh    = (const float*)d_in[2];  // [1536, 512]
  const float* bias   = (const float*)d_in[3];  // [1536]
  const float* bias_n = (const float*)d_in[4];  // [512]

  float*    hA      = (float*)d_ws;
  float*    hB      = hA + (size_t)BATCH * HSZ;
  unsigned* counter = (unsigned*)(hB + (size_t)BATCH * HSZ);

  const int hn = BATCH * HSZ;
  gru_init_kernel<<<(hn + 255) / 256, 256, 0, stream>>>(hA, counter, hn);

  gru_persistent_kernel<<<dim3(NBLK), dim3(256), 0, stream>>>(
      xs, wih, whh, bias, bias_n, hA, hB, counter, (float*)d_out);
}